// CD_49555332661898
// MI455X (gfx1250) — compile-verified
//
#include <hip/hip_runtime.h>

#define EPS 1e-8f
#define N_ 512
#define C_ 80
#define L_ 512
#define NTILE 16      // 512 / 32 tile rows
#define NPAIR 136     // #(ti<=tj) tile pairs
#define WPB 8         // waves per block (256 threads, wave32)

typedef __attribute__((ext_vector_type(2))) float v2f;
typedef __attribute__((ext_vector_type(4))) float v4f;
typedef __attribute__((ext_vector_type(8))) float v8f;

__device__ __forceinline__ float waveReduceSum(float v) {
#pragma unroll
    for (int o = 16; o > 0; o >>= 1) v += __shfl_xor(v, o, 32);
    return v;
}

// ---------------------------------------------------------------------------
// Kernel 1: per-(class,row) squared norms for student & teacher.
// One wave per (c,n); lanes read coalesced float4s.
// ---------------------------------------------------------------------------
__global__ void sqnorm_kernel(const float* __restrict__ S, const float* __restrict__ T,
                              float* __restrict__ sqS, float* __restrict__ sqT) {
    const int w    = blockIdx.x * WPB + (threadIdx.x >> 5);
    const int lane = threadIdx.x & 31;
    const int c = w % C_;
    const int n = w / C_;
    const size_t base = ((size_t)n * C_ + c) * L_;
    float ss = 0.f, st = 0.f;
#pragma unroll
    for (int i = 0; i < 4; ++i) {
        v4f a = *(const v4f*)(S + base + (size_t)(lane + i * 32) * 4);
        v4f b = *(const v4f*)(T + base + (size_t)(lane + i * 32) * 4);
        ss += a.x * a.x + a.y * a.y + a.z * a.z + a.w * a.w;
        st += b.x * b.x + b.y * b.y + b.z * b.z + b.w * b.w;
    }
    ss = waveReduceSum(ss);
    st = waveReduceSum(st);
    if (lane == 0) {
        sqS[c * N_ + n] = ss;
        sqT[c * N_ + n] = st;
    }
}

// ---------------------------------------------------------------------------
// Kernels 2 & 4: Gram + distance epilogue via V_WMMA_F32_16X16X4_F32.
// One wave owns a 32x32 output tile = 2x2 grid of 16x16 WMMA tiles,
// for BOTH student and teacher (8 f32 accumulators).
// Only tiles with tj>=ti are computed; tj>ti contributes twice (symmetry).
// PHASE 1: masked distance sums (for per-class means).
// PHASE 2: masked Huber(dsn - dtn) sums.
// ---------------------------------------------------------------------------
template <int PHASE>
__global__ void gram_kernel(const float* __restrict__ S, const float* __restrict__ T,
                            const int* __restrict__ tgt,
                            const float* __restrict__ sqS, const float* __restrict__ sqT,
                            const float* __restrict__ invS, const float* __restrict__ invT,
                            float* __restrict__ outA, float* __restrict__ outB) {
    const int w    = blockIdx.x * WPB + (threadIdx.x >> 5);
    const int lane = threadIdx.x & 31;
    const int c  = w / NPAIR;
    const int p0 = w % NPAIR;

    // map linear pair index -> (ti, tj) with tj >= ti
    int p = p0, ti = 0;
    while (p >= NTILE - ti) { p -= NTILE - ti; ++ti; }
    const int tj = ti + p;
    const int i0 = ti * 32, j0 = tj * 32;

    // WMMA f32 16x16x4 operand addressing:
    // A: lane m=lane&15 holds row (i0+m), VGPR0/1 = K = kb, kb+1, kb = 2*(lane>>4)
    // B (Fj^T): same per-lane pattern with row (j0 + lane&15)
    const int m  = lane & 15;
    const int kb = (lane >> 4) << 1;
    const size_t rowStride = (size_t)C_ * L_;           // stride between sample rows
    const size_t cbase     = (size_t)c * L_;

    const float* pAS0 = S + cbase + (size_t)(i0 + m) * rowStride + kb;
    const float* pAS1 = pAS0 + 16 * rowStride;
    const float* pBS0 = S + cbase + (size_t)(j0 + m) * rowStride + kb;
    const float* pBS1 = pBS0 + 16 * rowStride;
    const float* pAT0 = T + cbase + (size_t)(i0 + m) * rowStride + kb;
    const float* pAT1 = pAT0 + 16 * rowStride;
    const float* pBT0 = T + cbase + (size_t)(j0 + m) * rowStride + kb;
    const float* pBT1 = pBT0 + 16 * rowStride;

    v8f s00 = {}, s01 = {}, s10 = {}, s11 = {};
    v8f t00 = {}, t01 = {}, t10 = {}, t11 = {};

#pragma unroll 4
    for (int k = 0; k < L_; k += 4) {
        v2f aS0 = *(const v2f*)(pAS0 + k);
        v2f aS1 = *(const v2f*)(pAS1 + k);
        v2f bS0 = *(const v2f*)(pBS0 + k);
        v2f bS1 = *(const v2f*)(pBS1 + k);
        v2f aT0 = *(const v2f*)(pAT0 + k);
        v2f aT1 = *(const v2f*)(pAT1 + k);
        v2f bT0 = *(const v2f*)(pBT0 + k);
        v2f bT1 = *(const v2f*)(pBT1 + k);
        s00 = __builtin_amdgcn_wmma_f32_16x16x4_f32(false, aS0, false, bS0, (short)0, s00, false, false);
        s01 = __builtin_amdgcn_wmma_f32_16x16x4_f32(false, aS0, false, bS1, (short)0, s01, false, false);
        s10 = __builtin_amdgcn_wmma_f32_16x16x4_f32(false, aS1, false, bS0, (short)0, s10, false, false);
        s11 = __builtin_amdgcn_wmma_f32_16x16x4_f32(false, aS1, false, bS1, (short)0, s11, false, false);
        t00 = __builtin_amdgcn_wmma_f32_16x16x4_f32(false, aT0, false, bT0, (short)0, t00, false, false);
        t01 = __builtin_amdgcn_wmma_f32_16x16x4_f32(false, aT0, false, bT1, (short)0, t01, false, false);
        t10 = __builtin_amdgcn_wmma_f32_16x16x4_f32(false, aT1, false, bT0, (short)0, t10, false, false);
        t11 = __builtin_amdgcn_wmma_f32_16x16x4_f32(false, aT1, false, bT1, (short)0, t11, false, false);
    }

    v8f aS[2][2] = {{s00, s01}, {s10, s11}};
    v8f aT[2][2] = {{t00, t01}, {t10, t11}};

    // Epilogue: C/D layout -> element (M,N): N = lane&15, M = v + 8*(lane>>4)
    const float fac   = (ti == tj) ? 1.f : 2.f;
    const int giBase  = i0 + ((lane >> 4) << 3);
    const int gjBase  = j0 + (lane & 15);
    const float* sqSc = sqS + c * N_;
    const float* sqTc = sqT + c * N_;
    float is_ = 0.f, it_ = 0.f;
    if (PHASE == 2) { is_ = invS[c]; it_ = invT[c]; }

    float acc1 = 0.f, acc2 = 0.f;
#pragma unroll
    for (int si = 0; si < 2; ++si) {
#pragma unroll
        for (int sj = 0; sj < 2; ++sj) {
            const int gj = gjBase + sj * 16;
            const int mj = tgt[gj * C_ + c];
            const float sqjS = sqSc[gj];
            const float sqjT = sqTc[gj];
#pragma unroll
            for (int v = 0; v < 8; ++v) {
                const int gi = giBase + si * 16 + v;
                const int mi = tgt[gi * C_ + c];
                const float msk = (float)(mi & mj);
                const float gS = aS[si][sj][v];
                const float gT = aT[si][sj][v];
                const float d2s = fmaxf(sqSc[gi] + sqjS - 2.f * gS, EPS);
                const float d2t = fmaxf(sqTc[gi] + sqjT - 2.f * gT, EPS);
                const float ds = (gi == gj) ? 0.f : sqrtf(d2s);
                const float dt = (gi == gj) ? 0.f : sqrtf(d2t);
                if (PHASE == 1) {
                    acc1 += ds * msk;
                    acc2 += dt * msk;
                } else {
                    const float diff = ds * is_ - dt * it_;
                    const float ad = fabsf(diff);
                    const float hub = (ad < 1.f) ? 0.5f * diff * diff : ad - 0.5f;
                    acc1 += hub * msk;
                }
            }
        }
    }

    acc1 = waveReduceSum(acc1) * fac;
    if (PHASE == 1) acc2 = waveReduceSum(acc2) * fac;
    if (lane == 0) {
        const int slot = c * NPAIR + p0;
        outA[slot] = acc1;
        if (PHASE == 1) outB[slot] = acc2;
    }
}

// ---------------------------------------------------------------------------
// Kernel 3: per-class reduction -> reciprocal mean normalizers + positive count.
// ---------------------------------------------------------------------------
__global__ void reduce1_kernel(const float* __restrict__ partS, const float* __restrict__ partT,
                               const int* __restrict__ tgt,
                               float* __restrict__ invS, float* __restrict__ invT,
                               float* __restrict__ npos) {
    const int c = blockIdx.x;
    const int t = threadIdx.x; // 256 threads
    __shared__ float ls[256], lt[256], ln[256];
    ls[t] = (t < NPAIR) ? partS[c * NPAIR + t] : 0.f;
    lt[t] = (t < NPAIR) ? partT[c * NPAIR + t] : 0.f;
    ln[t] = (float)(tgt[t * C_ + c] + tgt[(t + 256) * C_ + c]);
    __syncthreads();
    for (int o = 128; o > 0; o >>= 1) {
        if (t < o) { ls[t] += ls[t + o]; lt[t] += lt[t + o]; ln[t] += ln[t + o]; }
        __syncthreads();
    }
    if (t == 0) {
        const float n = ln[0];
        const float cnt = fmaxf(n * (n - 1.f), 1.f);
        invS[c] = 1.f / fmaxf(ls[0] / cnt, EPS);
        invT[c] = 1.f / fmaxf(lt[0] / cnt, EPS);
        npos[c] = n;
    }
}

// ---------------------------------------------------------------------------
// Kernel 5: final scalar: sum_c [n_c>1] * hsum_c / max(n_c^2, 1)
// ---------------------------------------------------------------------------
__global__ void final_kernel(const float* __restrict__ partH, const float* __restrict__ npos,
                             float* __restrict__ out) {
    __shared__ float red[128];
    const int t = threadIdx.x; // 128 threads
    float term = 0.f;
    if (t < C_) {
        float s = 0.f;
        for (int i = 0; i < NPAIR; ++i) s += partH[t * NPAIR + i];
        const float n = npos[t];
        term = (n > 1.f) ? s / fmaxf(n * n, 1.f) : 0.f;
    }
    red[t] = term;
    __syncthreads();
    for (int o = 64; o > 0; o >>= 1) {
        if (t < o) red[t] += red[t + o];
        __syncthreads();
    }
    if (t == 0) out[0] = red[0];
}

// ---------------------------------------------------------------------------
extern "C" void kernel_launch(void* const* d_in, const int* in_sizes, int n_in,
                              void* d_out, int out_size, void* d_ws, size_t ws_size,
                              hipStream_t stream) {
    const float* S   = (const float*)d_in[0];
    const float* T   = (const float*)d_in[1];
    const int*   tgt = (const int*)d_in[2];

    float* ws    = (float*)d_ws;
    float* sqS   = ws;                     // 80*512
    float* sqT   = sqS + C_ * N_;          // 80*512
    float* partS = sqT + C_ * N_;          // 80*136
    float* partT = partS + C_ * NPAIR;     // 80*136
    float* partH = partT + C_ * NPAIR;     // 80*136
    float* invS  = partH + C_ * NPAIR;     // 80
    float* invT  = invS + C_;              // 80
    float* npos  = invT + C_;              // 80

    const int gramBlocks = (C_ * NPAIR) / WPB;               // 10880/8 = 1360
    sqnorm_kernel<<<(C_ * N_) / WPB, 256, 0, stream>>>(S, T, sqS, sqT);
    gram_kernel<1><<<gramBlocks, 256, 0, stream>>>(S, T, tgt, sqS, sqT, nullptr, nullptr, partS, partT);
    reduce1_kernel<<<C_, 256, 0, stream>>>(partS, partT, tgt, invS, invT, npos);
    gram_kernel<2><<<gramBlocks, 256, 0, stream>>>(S, T, tgt, sqS, sqT, invS, invT, partH, nullptr);
    final_kernel<<<1, 128, 0, stream>>>(partH, npos, (float*)d_out);
}